// PairwiseChannelConv_54709293416527
// MI455X (gfx1250) — compile-verified
//
#include <hip/hip_runtime.h>

typedef __attribute__((ext_vector_type(2))) float v2f;
typedef __attribute__((ext_vector_type(8))) float v8f;

// Bounds-handled x load: clamp address (always in-bounds load), select 0 for
// out-of-range taps via cndmask -- no EXEC divergence (WMMA needs EXEC==~0).
__device__ __forceinline__ float ldx(const float* __restrict__ xr, int w) {
    int wc = w < 0 ? 0 : (w > 1023 ? 1023 : w);
    float v = xr[wc];
    return (w == wc) ? v : 0.0f;
}

// One wave = (n, 16-wide w tile, 16-channel half).
// Conv7 along W expressed as two chained V_WMMA_F32_16X16X4_F32 (K=8, tap7=0).
__global__ __launch_bounds__(256)
void PairwiseChannelConv_kernel(const float* __restrict__ x,
                                const float* __restrict__ Wf,
                                float* __restrict__ out) {
    const int lane = threadIdx.x & 31;
    const int wave = blockIdx.x * 8 + (threadIdx.x >> 5);   // 1024 waves total

    const int ch = wave & 1;          // channel half: c0 = 0 or 16
    const int wt = (wave >> 1) & 63;  // w tile: 64 tiles of 16
    const int n  = wave >> 7;         // batch 0..7

    const int c0 = ch << 4;
    const int w0 = wt << 4;
    const int M  = lane & 15;         // A-matrix row / C-matrix col index
    const int hi = lane >> 4;         // 0: lanes 0-15, 1: lanes 16-31
    const int kb = hi ? 2 : 0;        // A/B K-pair base per ISA layout

    // ---- B matrices (filter, constant over h) -------------------------------
    // B 4x16 f32 layout (mirrors A): VGPR j holds row K = kb + j, col N = lane%16.
    // W is HWIO flat: Wf[k*32 + c].
    v2f b1, b2;
    b1.x = Wf[(kb + 0) * 32 + c0 + M];
    b1.y = Wf[(kb + 1) * 32 + c0 + M];
    b2.x = Wf[(kb + 4) * 32 + c0 + M];
    b2.y = hi ? 0.0f : Wf[5 * 32 + c0 + M];   // tap 7 (kb+5==7 when hi) is zero pad

    // ---- Pass 1: conv via WMMA + running top-2 over h -----------------------
    float m1[8], m2[8];
    int   am[8];
#pragma unroll
    for (int r = 0; r < 8; ++r) { m1[r] = -__builtin_inff(); m2[r] = -__builtin_inff(); am[r] = 0; }

    const int wbase = w0 + M + kb - 3;  // x index of tap kb for this lane's row

    for (int h = 0; h < 160; ++h) {
        const float* xr = x + (n * 160 + h) * 1024;
        __builtin_prefetch(xr + 1024, 0, 1);  // next h row -> global_prefetch_b8

        v2f a1, a2;
        a1.x = ldx(xr, wbase + 0);                  // tap kb
        a1.y = ldx(xr, wbase + 1);                  // tap kb+1
        a2.x = ldx(xr, wbase + 4);                  // tap kb+4
        a2.y = hi ? 0.0f : ldx(xr, wbase + 5);      // tap kb+5 (==7 -> 0)

        v8f d = {};
        d = __builtin_amdgcn_wmma_f32_16x16x4_f32(false, a2, false, b2, (short)0, d, false, false);
        d = __builtin_amdgcn_wmma_f32_16x16x4_f32(false, a1, false, b1, (short)0, d, false, false);

#pragma unroll
        for (int r = 0; r < 8; ++r) {
            float v = d[r];
            bool top = v > m1[r];
            float nm2 = top ? m1[r] : ((v > m2[r]) ? v : m2[r]);
            m1[r] = top ? v : m1[r];
            am[r] = top ? h : am[r];
            m2[r] = nm2;
        }
    }

    // ---- Pass 2: recompute conv tile, add pairwise term, stream out ---------
    for (int h = 0; h < 160; ++h) {
        const float* xr = x + (n * 160 + h) * 1024;

        v2f a1, a2;
        a1.x = ldx(xr, wbase + 0);
        a1.y = ldx(xr, wbase + 1);
        a2.x = ldx(xr, wbase + 4);
        a2.y = hi ? 0.0f : ldx(xr, wbase + 5);

        v8f d = {};
        d = __builtin_amdgcn_wmma_f32_16x16x4_f32(false, a2, false, b2, (short)0, d, false, false);
        d = __builtin_amdgcn_wmma_f32_16x16x4_f32(false, a1, false, b1, (short)0, d, false, false);

        // C/D layout: VGPR r, lanes 0-15 -> row M=r; lanes 16-31 -> row M=r+8.
        // row M = w offset in tile; col N = lane%16 = channel offset.
        // Output is a 168MB write-once stream: use non-temporal stores so it
        // doesn't evict the L2-resident x working set (write-BW-bound kernel).
        float* orow = out + (((long)(n * 160 + h) * 1024) + w0) * 32 + c0 + M;
#pragma unroll
        for (int r = 0; r < 8; ++r) {
            float v = d[r];
            float other = (h == am[r]) ? m2[r] : m1[r];
            int wrow = r + (hi << 3);
            __builtin_nontemporal_store(v + other, orow + (long)wrow * 32);
        }
    }
}

extern "C" void kernel_launch(void* const* d_in, const int* in_sizes, int n_in,
                              void* d_out, int out_size, void* d_ws, size_t ws_size,
                              hipStream_t stream) {
    (void)in_sizes; (void)n_in; (void)d_ws; (void)ws_size; (void)out_size;
    const float* x  = (const float*)d_in[0];   // (8,160,1024,1) f32
    const float* Wf = (const float*)d_in[1];   // (1,7,1,32)    f32
    float* out = (float*)d_out;                // (8,160,1024,32) f32

    // 1024 waves = 8 n * 64 w-tiles * 2 channel halves; 8 waves per block.
    PairwiseChannelConv_kernel<<<dim3(128), dim3(256), 0, stream>>>(x, Wf, out);
}